// Body_72086731096465
// MI455X (gfx1250) — compile-verified
//
#include <hip/hip_runtime.h>
#include <math.h>

// ---------------- problem constants (match reference) ----------------
#define Nn 20000
#define Ee 320000
#define Gg 64
#define DX 128
#define DH 128
#define DE 64
#define NTILES (Nn / 16)   // 1250 16-node tiles

typedef float v2f __attribute__((ext_vector_type(2)));
typedef float v8f __attribute__((ext_vector_type(8)));

// ---------------- WMMA f32 16x16x4 helper ----------------
// D(16x16,f32) = A(16x4,f32) * B(4x16,f32) + C
// A frag: lane l -> row m=l&15, K pair {2g,2g+1}, g=l>>4
// B frag: lane l -> col n=l&15, rows {k0+2g, k0+2g+1}
// C frag: vgpr j, lane l -> row j+8g, col n=l&15
__device__ __forceinline__ v8f wmma_f32(v2f a, v2f b, v8f c) {
    return __builtin_amdgcn_wmma_f32_16x16x4_f32(
        false, a, false, b, (short)0, c, false, false);
}

// ---------------- async global -> LDS staging (ASYNCcnt path) ----------------
// Each lane issues global_load_async_to_lds_b128: 16 bytes memory -> LDS at the
// per-lane LDS byte offset (low 32 bits of the generic LDS address).
__device__ __forceinline__ void asyncCopyToLds(float* ldsDst, const float* gsrc,
                                               int nFloats, int tid, int nthr) {
    unsigned base = (unsigned)(uintptr_t)ldsDst;   // LDS byte offset (addr[31:0])
    for (int i = tid; i < (nFloats >> 2); i += nthr) {
        unsigned loff = base + i * 16;
        const float* g = gsrc + i * 4;
        asm volatile("global_load_async_to_lds_b128 %0, %1, off"
                     :: "v"(loff), "v"(g) : "memory");
    }
}
__device__ __forceinline__ void waitAsync0() {
    asm volatile("s_wait_asynccnt 0x0" ::: "memory");
}

// ---------------- ordered-float atomic-max keys ----------------
__device__ __forceinline__ unsigned ordKey(float f) {
    unsigned u = __float_as_uint(f);
    return (u & 0x80000000u) ? ~u : (u | 0x80000000u);
}
__device__ __forceinline__ float ordVal(unsigned k) {
    unsigned u = (k & 0x80000000u) ? (k & 0x7fffffffu) : ~k;
    return __uint_as_float(u);
}

__device__ __forceinline__ float waveReduceSum(float v) {
    v += __shfl_xor(v, 16, 32);
    v += __shfl_xor(v, 8, 32);
    v += __shfl_xor(v, 4, 32);
    v += __shfl_xor(v, 2, 32);
    v += __shfl_xor(v, 1, 32);
    return v;
}

__device__ __forceinline__ float sigmoidf(float v) {
    return 1.0f / (1.0f + expf(-v));
}

// ================================================================
// Kernel 1: P[n,:] = [x|h][n,:] @ W_emb[0:256,:] + b_emb   (WMMA)
// 4 waves/block, one 16-node tile per wave (64 nodes/block).
// Weight K-chunks (64 rows x 128 cols = 32KB) double-buffered in LDS
// via global_load_async_to_lds_b128, overlapped with WMMA compute.
// ================================================================
__global__ __launch_bounds__(128) void k_node_pre(
    const float* __restrict__ x, const float* __restrict__ h,
    const float* __restrict__ Wemb, const float* __restrict__ bemb,
    float* __restrict__ P)
{
    __shared__ float sW[2][64 * DH];   // 64KB double buffer

    const int wave = threadIdx.x >> 5;
    const int lane = threadIdx.x & 31;
    const int m = lane & 15;
    const int g = lane >> 4;
    const int tile = blockIdx.x * 4 + wave;
    const bool active = tile < NTILES;
    const int nbase = tile * 16;

    // prefetch chunk 0
    asyncCopyToLds(sW[0], Wemb, 64 * DH, threadIdx.x, 128);
    waitAsync0();
    __syncthreads();

    v8f acc[8] = {};
    const int NC = 4;                 // 4 chunks of 64 K-rows (K=256)
    for (int c = 0; c < NC; ++c) {
        if (c + 1 < NC)               // overlap next weight chunk with compute
            asyncCopyToLds(sW[(c + 1) & 1], Wemb + (size_t)(c + 1) * 64 * DH,
                           64 * DH, threadIdx.x, 128);
        if (active) {
            const int kglob = c * 64;
            const float* arow = (kglob < 128)
                ? (x + (size_t)(nbase + m) * DX + kglob)
                : (h + (size_t)(nbase + m) * DH + (kglob - 128));
            const float* wbuf = sW[c & 1];
            for (int k0 = 0; k0 < 64; k0 += 4) {
                v2f a; a.x = arow[k0 + 2 * g]; a.y = arow[k0 + 2 * g + 1];
                const float* wrow = wbuf + (k0 + 2 * g) * DH;
#pragma unroll
                for (int t = 0; t < 8; ++t) {
                    v2f b; b.x = wrow[t * 16 + m]; b.y = wrow[DH + t * 16 + m];
                    acc[t] = wmma_f32(a, b, acc[t]);
                }
            }
        }
        waitAsync0();
        __syncthreads();
    }

    if (active) {
#pragma unroll
        for (int t = 0; t < 8; ++t) {
            const int cc = t * 16 + m;
            const float bb = bemb[cc];
#pragma unroll
            for (int j = 0; j < 8; ++j) {
                const int row = nbase + j + 8 * g;
                P[(size_t)row * DH + cc] = acc[t][j] + bb;
            }
        }
    }
}

// ================================================================
// Kernel 2: per-node attention partials (one wave per node)
// ================================================================
__global__ __launch_bounds__(256) void k_node_scores(
    const float* __restrict__ x, const float* __restrict__ h,
    const float* __restrict__ wagg,
    float* __restrict__ s_src, float* __restrict__ s_dst)
{
    const int n = blockIdx.x * 8 + (threadIdx.x >> 5);
    const int lane = threadIdx.x & 31;
    float a = 0.f, b = 0.f;
    for (int c = lane; c < 128; c += 32) {
        const float xv = x[(size_t)n * DX + c];
        const float hv = h[(size_t)n * DH + c];
        a += xv * wagg[c]       + hv * wagg[128 + c];
        b += xv * wagg[256 + c] + hv * wagg[384 + c];
    }
    a = waveReduceSum(a);
    b = waveReduceSum(b);
    if (lane == 0) { s_src[n] = a; s_dst[n] = b; }
}

// ================================================================
// Kernel 3: att[e] = s_src[rol]+s_dst[col]+ea[e].w_agg[512:576]
//           + per-dst-node running max (ordered-uint atomicMax)
// ================================================================
__global__ __launch_bounds__(256) void k_edge_att(
    const float* __restrict__ ea, const float* __restrict__ wagg,
    const int* __restrict__ ei,
    const float* __restrict__ s_src, const float* __restrict__ s_dst,
    float* __restrict__ att, unsigned* __restrict__ mkey)
{
    const int e = blockIdx.x * 8 + (threadIdx.x >> 5);
    const int lane = threadIdx.x & 31;
    float s = ea[(size_t)e * DE + lane]      * wagg[512 + lane]
            + ea[(size_t)e * DE + 32 + lane] * wagg[544 + lane];
    s = waveReduceSum(s);
    if (lane == 0) {
        const int cn = ei[Ee + e];
        const float v = s + s_src[ei[e]] + s_dst[cn];
        att[e] = v;
        atomicMax(&mkey[cn], ordKey(v));
    }
}

// ================================================================
// Kernel 4: e_exp[e] = exp(att-m[col]);  z[col] += e_exp
// ================================================================
__global__ __launch_bounds__(256) void k_edge_exp(
    const int* __restrict__ ei, const float* __restrict__ att,
    const unsigned* __restrict__ mkey,
    float* __restrict__ e_exp, float* __restrict__ z)
{
    const int e = blockIdx.x * 256 + threadIdx.x;
    const int cn = ei[Ee + e];
    const float ex = expf(att[e] - ordVal(mkey[cn]));
    e_exp[e] = ex;
    atomicAdd(&z[cn], ex);
}

// ================================================================
// Kernel 5: edge embedding (WMMA) + softmax-weighted scatter
// emb[e,:] = relu(P[rol[e],:] + ea[e,:] @ W_emb[256:320,:])
// cand[col[e],:] += alpha[e] * emb[e,:]
// W2 (32KB) staged once per block via async-to-LDS.
// ================================================================
__global__ __launch_bounds__(128) void k_edge_emb_scatter(
    const float* __restrict__ P, const float* __restrict__ ea,
    const float* __restrict__ Wemb, const int* __restrict__ ei,
    const float* __restrict__ e_exp, const float* __restrict__ z,
    float* __restrict__ cand)
{
    __shared__ float sW[DE * DH]; // W_emb rows 256..319  (32KB)
    asyncCopyToLds(sW, Wemb + (size_t)256 * DH, DE * DH, threadIdx.x, 128);
    waitAsync0();
    __syncthreads();

    const int lane = threadIdx.x & 31;
    const int m = lane & 15;
    const int g = lane >> 4;
    const int ebase = (blockIdx.x * 4 + (threadIdx.x >> 5)) * 16;

    v8f acc[8] = {};
    const float* arow = ea + (size_t)(ebase + m) * DE;
    for (int k0 = 0; k0 < DE; k0 += 4) {
        v2f a; a.x = arow[k0 + 2 * g]; a.y = arow[k0 + 2 * g + 1];
        const float* wrow = sW + (k0 + 2 * g) * DH;
#pragma unroll
        for (int t = 0; t < 8; ++t) {
            v2f b; b.x = wrow[t * 16 + m]; b.y = wrow[DH + t * 16 + m];
            acc[t] = wmma_f32(a, b, acc[t]);
        }
    }
#pragma unroll
    for (int j = 0; j < 8; ++j) {
        const int e  = ebase + j + 8 * g;
        const int r  = ei[e];        // rol
        const int cn = ei[Ee + e];   // col
        const float al = e_exp[e] / (z[cn] + 1e-16f);
        const float* Pr = P + (size_t)r * DH;
        float* Cd = cand + (size_t)cn * DH;
#pragma unroll
        for (int t = 0; t < 8; ++t) {
            const int c = t * 16 + m;
            float v = acc[t][j] + Pr[c];
            v = v > 0.f ? v : 0.f;
            atomicAdd(Cd + c, v * al);
        }
    }
}

// ================================================================
// Kernel 6: gated update (WMMA): gate = sigmoid([h|cand|x]@W_upd+b)
// node_feat = gate*cand + (1-gate)*h  -> d_out[0 : N*DH)
// Same double-buffered async weight staging as kernel 1 (K=384, 6 chunks).
// ================================================================
__global__ __launch_bounds__(128) void k_update(
    const float* __restrict__ h, const float* __restrict__ cand,
    const float* __restrict__ x, const float* __restrict__ Wupd,
    const float* __restrict__ bupd, float* __restrict__ nf)
{
    __shared__ float sW[2][64 * DH];   // 64KB double buffer

    const int wave = threadIdx.x >> 5;
    const int lane = threadIdx.x & 31;
    const int m = lane & 15;
    const int g = lane >> 4;
    const int tile = blockIdx.x * 4 + wave;
    const bool active = tile < NTILES;
    const int nbase = tile * 16;

    asyncCopyToLds(sW[0], Wupd, 64 * DH, threadIdx.x, 128);
    waitAsync0();
    __syncthreads();

    v8f acc[8] = {};
    const int NC = 6;                 // 6 chunks of 64 K-rows (K=384)
    for (int c = 0; c < NC; ++c) {
        if (c + 1 < NC)
            asyncCopyToLds(sW[(c + 1) & 1], Wupd + (size_t)(c + 1) * 64 * DH,
                           64 * DH, threadIdx.x, 128);
        if (active) {
            const int kglob = c * 64;
            const float* arow;
            if (kglob < 128)      arow = h    + (size_t)(nbase + m) * DH + kglob;
            else if (kglob < 256) arow = cand + (size_t)(nbase + m) * DH + (kglob - 128);
            else                  arow = x    + (size_t)(nbase + m) * DX + (kglob - 256);
            const float* wbuf = sW[c & 1];
            for (int k0 = 0; k0 < 64; k0 += 4) {
                v2f a; a.x = arow[k0 + 2 * g]; a.y = arow[k0 + 2 * g + 1];
                const float* wrow = wbuf + (k0 + 2 * g) * DH;
#pragma unroll
                for (int t = 0; t < 8; ++t) {
                    v2f b; b.x = wrow[t * 16 + m]; b.y = wrow[DH + t * 16 + m];
                    acc[t] = wmma_f32(a, b, acc[t]);
                }
            }
        }
        waitAsync0();
        __syncthreads();
    }

    if (active) {
#pragma unroll
        for (int t = 0; t < 8; ++t) {
            const int cc = t * 16 + m;
            const float bb = bupd[cc];
#pragma unroll
            for (int j = 0; j < 8; ++j) {
                const int row = nbase + j + 8 * g;
                const float gate = sigmoidf(acc[t][j] + bb);
                const float cv = cand[(size_t)row * DH + cc];
                const float hv = h[(size_t)row * DH + cc];
                nf[(size_t)row * DH + cc] = gate * cv + (1.f - gate) * hv;
            }
        }
    }
}

// ================================================================
// Kernel 7: readout score per node + per-graph max
// ================================================================
__global__ __launch_bounds__(256) void k_ro(
    const float* __restrict__ nf, const float* __restrict__ x,
    const float* __restrict__ wro, const int* __restrict__ batch,
    float* __restrict__ ro, unsigned* __restrict__ gmkey)
{
    const int n = blockIdx.x * 8 + (threadIdx.x >> 5);
    const int lane = threadIdx.x & 31;
    float s = 0.f;
    for (int c = lane; c < 128; c += 32)
        s += nf[(size_t)n * DH + c] * wro[c] + x[(size_t)n * DX + c] * wro[128 + c];
    s = waveReduceSum(s);
    if (lane == 0) {
        ro[n] = s;
        atomicMax(&gmkey[batch[n]], ordKey(s));
    }
}

// ================================================================
// Kernel 8: en[n]=exp(ro-m[batch]);  gz[batch]+=en
// ================================================================
__global__ __launch_bounds__(256) void k_gexp(
    const float* __restrict__ ro, const unsigned* __restrict__ gmkey,
    const int* __restrict__ batch, float* __restrict__ en, float* __restrict__ gz)
{
    const int n = blockIdx.x * 256 + threadIdx.x;
    if (n >= Nn) return;
    const int g = batch[n];
    const float ex = expf(ro[n] - ordVal(gmkey[g]));
    en[n] = ex;
    atomicAdd(&gz[g], ex);
}

// ================================================================
// Kernel 9: graph_feat[g,:] += beta[n]*node_feat[n,:]
// ================================================================
__global__ __launch_bounds__(256) void k_gsum(
    const float* __restrict__ nf, const float* __restrict__ en,
    const float* __restrict__ gz, const int* __restrict__ batch,
    float* __restrict__ gfeat)
{
    const size_t idx = (size_t)blockIdx.x * 256 + threadIdx.x; // N*128 total
    const int n = (int)(idx >> 7);
    const int c = (int)(idx & 127);
    const int g = batch[n];
    const float beta = en[n] / (gz[g] + 1e-16f);
    atomicAdd(&gfeat[(size_t)g * DH + c], nf[idx] * beta);
}

// ================================================================
// Kernel 10: confidence[g] = sigmoid(graph_feat[g].W_score + b)
// ================================================================
__global__ __launch_bounds__(32) void k_conf(
    const float* __restrict__ gfeat, const float* __restrict__ Wscore,
    const float* __restrict__ bscore, float* __restrict__ conf)
{
    const int g = blockIdx.x;
    const int lane = threadIdx.x;
    float s = 0.f;
    for (int c = lane; c < 128; c += 32)
        s += gfeat[(size_t)g * DH + c] * Wscore[c];
    s = waveReduceSum(s);
    if (lane == 0) conf[g] = sigmoidf(s + bscore[0]);
}

// ================================================================
extern "C" void kernel_launch(void* const* d_in, const int* in_sizes, int n_in,
                              void* d_out, int out_size, void* d_ws, size_t ws_size,
                              hipStream_t stream) {
    const float* x    = (const float*)d_in[0];
    const float* h    = (const float*)d_in[1];
    const float* ea   = (const float*)d_in[2];
    const float* Wemb = (const float*)d_in[3];
    const float* bemb = (const float*)d_in[4];
    const float* wagg = (const float*)d_in[5];
    const float* Wupd = (const float*)d_in[6];
    const float* bupd = (const float*)d_in[7];
    const float* wro  = (const float*)d_in[8];
    const float* Wsc  = (const float*)d_in[9];
    const float* bsc  = (const float*)d_in[10];
    const int*   ei   = (const int*)d_in[11];   // [2,E]
    const int*   batch= (const int*)d_in[12];   // [N]

    float* nf   = (float*)d_out;                // node_feat  [N,128]
    float* conf = nf + (size_t)Nn * DH;         // confidence [G]

    // -------- workspace layout --------
    float* ws    = (float*)d_ws;
    float* P     = ws;                           // N*128
    float* s_src = P + (size_t)Nn * DH;          // N
    float* s_dst = s_src + Nn;                   // N
    float* att   = s_dst + Nn;                   // E
    float* e_exp = att + Ee;                     // E
    float* ro    = e_exp + Ee;                   // N
    float* en    = ro + Nn;                      // N
    // zero-initialized region (contiguous, one memset):
    unsigned* mkey  = (unsigned*)(en + Nn);      // N   (key 0 == -inf identity)
    float*    z     = (float*)(mkey + Nn);       // N
    float*    cand  = z + Nn;                    // N*128
    unsigned* gmkey = (unsigned*)(cand + (size_t)Nn * DH); // G
    float*    gz    = (float*)(gmkey + Gg);      // G
    float*    gfeat = gz + Gg;                   // G*128
    const size_t zeroBytes =
        ((size_t)Nn * 2 + (size_t)Nn * DH + Gg * 2 + (size_t)Gg * DH) * sizeof(float);
    hipMemsetAsync(mkey, 0, zeroBytes, stream);  // graph-capture safe

    // -------- pipeline --------
    k_node_pre        <<<(NTILES + 3) / 4, 128, 0, stream>>>(x, h, Wemb, bemb, P);
    k_node_scores     <<<Nn / 8, 256,  0, stream>>>(x, h, wagg, s_src, s_dst);
    k_edge_att        <<<Ee / 8, 256,  0, stream>>>(ea, wagg, ei, s_src, s_dst, att, mkey);
    k_edge_exp        <<<Ee / 256, 256,0, stream>>>(ei, att, mkey, e_exp, z);
    k_edge_emb_scatter<<<Ee / 64, 128, 0, stream>>>(P, ea, Wemb, ei, e_exp, z, cand);
    k_update          <<<(NTILES + 3) / 4, 128, 0, stream>>>(h, cand, x, Wupd, bupd, nf);
    k_ro              <<<Nn / 8, 256,  0, stream>>>(nf, x, wro, batch, ro, gmkey);
    k_gexp            <<<(Nn + 255) / 256, 256, 0, stream>>>(ro, gmkey, batch, en, gz);
    k_gsum            <<<((size_t)Nn * DH) / 256, 256, 0, stream>>>(nf, en, gz, batch, gfeat);
    k_conf            <<<Gg, 32, 0, stream>>>(gfeat, Wsc, bsc, conf);
}